// VectorQuantizer_63170378990323
// MI455X (gfx1250) — compile-verified
//
#include <hip/hip_runtime.h>
#include <math.h>

// ---------------- problem constants ----------------
#define N_TOK   131072          // 32 * 64 * 64 tokens
#define N_EMB   512
#define E_DIM   64
#define HW      4096            // 64*64 spatial
#define N_MTILE (N_TOK / 16)    // 8192 M-tiles of 16 rows

// d_out flat layout (floats), reference tuple order:
// [loss][z_q 8388608][perplexity][one_hot 67108864][idx 131072]
#define ZQ_OFF  ((size_t)1)
#define PPL_OFF ((size_t)8388609)
#define ENC_OFF ((size_t)8388610)
#define IDX_OFF ((size_t)75497474)

typedef __attribute__((ext_vector_type(2))) float v2f;
typedef __attribute__((ext_vector_type(8))) float v8f;

// dynamic LDS layout (floats): [codebook 32768][e2 512][per-wave idx 8*16 ints]
#define SM_E2     (N_EMB * E_DIM)
#define SM_IDX    (N_EMB * E_DIM + N_EMB)
#define SM_FLOATS (N_EMB * E_DIM + N_EMB + 8 * 16)

// ---------------- kernel 1: zero one-hot + hist, precompute ||e||^2 ----------
// one-hot region byte offset is 8-aligned (not 16), so use float2 stores.
__global__ void vq_fill(float2* __restrict__ enc, unsigned* __restrict__ hist,
                        const float* __restrict__ e, float* __restrict__ e2) {
    size_t gid = (size_t)blockIdx.x * blockDim.x + threadIdx.x;   // 33554432 float2
    float2 zz; zz.x = 0.0f; zz.y = 0.0f;
    enc[gid] = zz;
    if (gid < N_EMB) {
        hist[gid] = 0u;
        float s = 0.0f;
#pragma unroll 4
        for (int k = 0; k < E_DIM; ++k) {
            const float v = e[gid * E_DIM + k];
            s += v * v;
        }
        e2[gid] = s;
    }
}

// ---------------- kernel 2: WMMA distance + argmin + scatter ----------------
// One wave per 16-row M-tile; codebook staged in LDS, shared by 8 waves.
// fp32 WMMA 16x16x4 layouts:
//   A (16x4):  lanes 0-15 rows 0-15, VGPR0/1 = K(0,1); lanes 16-31 = K(2,3)
//   B (4x16):  lanes 0-15 cols 0-15, VGPR0/1 = K(0,1); lanes 16-31 = K(2,3)
//   C/D f32:   VGPR v, lanes 0-15 -> row v, lanes 16-31 -> row v+8; col = lane%16
__global__ void __launch_bounds__(256)
vq_main(const float* __restrict__ z, const float* __restrict__ e,
        const float* __restrict__ e2g,
        float* __restrict__ out, float* __restrict__ loss_part,
        unsigned* __restrict__ hist) {
    extern __shared__ float smem[];
    float* se   = smem;                 // [512][64] codebook
    float* se2  = smem + SM_E2;         // [512]    ||e||^2
    int*   sidx = (int*)(smem + SM_IDX);// [8][16]  per-wave row argmin

    // stage codebook + norms into LDS (128KB + 2KB), 8 waves cooperate
    {
        const float4* src = (const float4*)e;
        float4* dst = (float4*)se;
        for (int i = threadIdx.x; i < (N_EMB * E_DIM) / 4; i += 256) dst[i] = src[i];
        for (int i = threadIdx.x; i < N_EMB; i += 256) se2[i] = e2g[i];
    }
    __syncthreads();

    const int lane  = threadIdx.x & 31;
    const int wave  = threadIdx.x >> 5;
    const int tile  = blockIdx.x * 8 + wave;
    const int mbase = tile * 16;
    const int lrow  = lane & 15;
    const int khalf = lane >> 4;

    // preload A fragments: z is [B,C,H,W] -> z[m][k] = z[(b*64+k)*4096 + hw]
    const int m = mbase + lrow;
    const size_t zrb = (size_t)(m >> 12) * (E_DIM * HW) + (size_t)(m & (HW - 1));
    v2f A[16];
#pragma unroll
    for (int s = 0; s < 16; ++s) {
        const int k = 4 * s + 2 * khalf;
        A[s].x = z[zrb + (size_t)k * HW];
        A[s].y = z[zrb + (size_t)(k + 1) * HW];
    }

    float minval[8];
    int   minidx[8];
#pragma unroll
    for (int v = 0; v < 8; ++v) { minval[v] = 3.4e38f; minidx[v] = 0; }

    for (int j = 0; j < N_EMB / 16; ++j) {
        const int   col  = j * 16 + lrow;
        const float* erow = se + (size_t)col * E_DIM;
        v8f acc = {};
#pragma unroll
        for (int s = 0; s < 16; ++s) {
            const int k = 4 * s + 2 * khalf;
            v2f B;
            B.x = erow[k];               // ds_load_b64 (8B aligned: k even)
            B.y = erow[k + 1];
            acc = __builtin_amdgcn_wmma_f32_16x16x4_f32(
                      false, A[s], false, B, (short)0, acc, false, false);
        }
        const float e2 = se2[col];
#pragma unroll
        for (int v = 0; v < 8; ++v) {
            const float d = e2 - 2.0f * acc[v];   // ||z||^2 const per row: dropped
            if (d < minval[v]) { minval[v] = d; minidx[v] = col; }
        }
    }

    // argmin across the 16 lanes of each half (lexicographic -> first-min tiebreak)
#pragma unroll
    for (int v = 0; v < 8; ++v) {
#pragma unroll
        for (int off = 1; off <= 8; off <<= 1) {
            const float ov = __shfl_xor(minval[v], off, 32);
            const int   oi = __shfl_xor(minidx[v], off, 32);
            if (ov < minval[v] || (ov == minval[v] && oi < minidx[v])) {
                minval[v] = ov; minidx[v] = oi;
            }
        }
    }

    // publish per-row argmin so every lane can pick up its row's index
#pragma unroll
    for (int v = 0; v < 8; ++v) {
        if (lane == 0)  sidx[wave * 16 + v]     = minidx[v];   // rows 0..7
        if (lane == 16) sidx[wave * 16 + 8 + v] = minidx[v];   // rows 8..15
    }
    __syncthreads();

    float* __restrict__ zq   = out + ZQ_OFF;
    float* __restrict__ enc  = out + ENC_OFF;
    float* __restrict__ idxo = out + IDX_OFF;

    // coalesced gather/scatter: lane -> (row = lane&15, channel parity = lane>>4)
    const int myrow = lane & 15;
    const int cpair = lane >> 4;
    const int myidx = sidx[wave * 16 + myrow];
    const int ng    = mbase + myrow;
    const size_t rb = (size_t)(ng >> 12) * (E_DIM * HW) + (size_t)(ng & (HW - 1));

    float lsum = 0.0f;
#pragma unroll
    for (int c0 = 0; c0 < E_DIM; c0 += 2) {
        const int c = c0 + cpair;
        const float ev = se[myidx * E_DIM + c];
        const float zv = z[rb + (size_t)c * HW];
        zq[rb + (size_t)c * HW] = ev;               // straight-through fwd value
        const float dd = ev - zv;
        lsum += dd * dd;
    }
    if (lane < 16) {
        idxo[ng] = (float)myidx;
        enc[(size_t)ng * N_EMB + myidx] = 1.0f;
        atomicAdd(&hist[myidx], 1u);                // integer atomics: deterministic
    }

#pragma unroll
    for (int off = 1; off <= 16; off <<= 1) lsum += __shfl_xor(lsum, off, 32);
    if (lane == 0) loss_part[tile] = lsum;
}

// ---------------- kernel 3: deterministic reductions ----------------
__global__ void vq_final(const float* __restrict__ loss_part,
                         const unsigned* __restrict__ hist,
                         float* __restrict__ out) {
    __shared__ float s1[256];
    __shared__ float s2[256];
    const int t = threadIdx.x;
    float a = 0.0f;
    for (int i = t; i < N_MTILE; i += 256) a += loss_part[i];
    float p = 0.0f;
    for (int i = t; i < N_EMB; i += 256) {
        const float em = (float)hist[i] * (1.0f / (float)N_TOK);
        p -= em * logf(em + 1e-10f);
    }
    s1[t] = a; s2[t] = p;
    __syncthreads();
    for (int stw = 128; stw > 0; stw >>= 1) {
        if (t < stw) { s1[t] += s1[t + stw]; s2[t] += s2[t + stw]; }
        __syncthreads();
    }
    if (t == 0) {
        out[0]       = 1.25f * s1[0] / 8388608.0f;  // (1 + commit_cost) * mean(diff^2)
        out[PPL_OFF] = expf(s2[0]);
    }
}

// ---------------- launch ----------------
extern "C" void kernel_launch(void* const* d_in, const int* in_sizes, int n_in,
                              void* d_out, int out_size, void* d_ws, size_t ws_size,
                              hipStream_t stream) {
    (void)in_sizes; (void)n_in; (void)out_size; (void)ws_size;
    const float* z = (const float*)d_in[0];   // [32,64,64,64] fp32
    const float* e = (const float*)d_in[1];   // [512,64] fp32
    float*    out       = (float*)d_out;
    float*    loss_part = (float*)d_ws;                          // 8192 floats
    unsigned* hist      = (unsigned*)((char*)d_ws + 32768);      // 512 u32
    float*    e2        = (float*)((char*)d_ws + 34816);         // 512 f32

    // zero 67,108,864 floats (33,554,432 float2) + hist, precompute ||e||^2
    vq_fill<<<131072, 256, 0, stream>>>((float2*)(out + ENC_OFF), hist, e, e2);
    // 1024 blocks * 8 waves = 8192 M-tiles; 130.5 KB dynamic LDS per block
    vq_main<<<1024, 256, SM_FLOATS * sizeof(float), stream>>>(
        z, e, e2, out, loss_part, hist);
    vq_final<<<1, 256, 0, stream>>>(loss_part, hist, out);
}